// ImageCaptioningModel_81295140978991
// MI455X (gfx1250) — compile-verified
//
#include <hip/hip_runtime.h>

#define BATCH     512
#define VOCAB     50257
#define PAST_LEN  64
#define NTHREADS  512
#define NWAVES    (NTHREADS / 32)
#define CWORDS    ((VOCAB + 3) / 4)      // byte-packed counts, u32 words
#define POOL_QW   6283                   // 50264 B: max(counts 50260 B, bins 16384 B)

typedef __attribute__((ext_vector_type(2))) float v2f;
typedef __attribute__((ext_vector_type(4))) float v4f;
typedef __attribute__((ext_vector_type(8))) float v8f;

// floor(p * 2^32) computed exactly from the bit pattern of a positive float p<=1.
// Pure integer ops -- replaces the f64 cvt/trunc/floor chain in the hot loops.
__device__ __forceinline__ unsigned long long fix_from_key(unsigned int k) {
  unsigned int e = k >> 23;                                  // sign bit is 0
  unsigned long long mant = (unsigned long long)((k & 0x7FFFFFu) | 0x800000u);
  if (e >= 118) return mant << (e - 118);                    // e<=127 -> shift<=9
  if (e >= 55)  return mant >> (118u - e);                   // shift 1..63
  return 0ull;                                               // underflows to 0
}

// 3-level radix select over the float bit pattern of p = exp(x - rowM) (p > 0 so
// raw bits are monotonic). Finds the first element, in descending-prob order,
// whose inclusive fixed-point cumulative sum crosses T (>= if !strict, > if strict).
// Returns its exact 32-bit key; *outAcc = mass strictly above it, *outBinSum = mass
// exactly at that key. Deterministic: associative u64 fixed-point LDS atomics.
__device__ __forceinline__ unsigned int radix_select(
    const float* __restrict__ xrow, float rowM,
    int head, int nvec, int tailStart,
    unsigned long long T, bool strict,
    unsigned long long* bins,
    unsigned int* sh_prefix, unsigned long long* sh_acc, unsigned long long* sh_bin,
    unsigned long long* outAcc, unsigned long long* outBinSum)
{
  const int tid = threadIdx.x;
  unsigned int prefix = 0;
  int pb = 0;
  unsigned long long acc = 0ull, chosen = 0ull;
  const v4f* xv = (const v4f*)(xrow + head);
  #pragma unroll 1
  for (int lev = 0; lev < 3; ++lev) {
    const int bits = (lev < 2) ? 11 : 10;                    // 11+11+10 = 32 bits
    const int nb = 1 << bits;
    const int shift = 32 - pb - bits;
    for (int i = tid; i < nb; i += NTHREADS) bins[i] = 0ull;
    __syncthreads();
    auto accum = [&](float x) {
      float p = expf(x - rowM);
      unsigned int k = __float_as_uint(p);
      if (pb == 0 || (k >> (32 - pb)) == prefix)
        atomicAdd(&bins[(k >> shift) & (unsigned)(nb - 1)], fix_from_key(k));
    };
    for (int v = tid; v < head; v += NTHREADS) accum(xrow[v]);
    for (int i = tid; i < nvec; i += NTHREADS) {
      v4f x4 = xv[i];                                        // aligned b128, L2-hot
      accum(x4.x); accum(x4.y); accum(x4.z); accum(x4.w);
    }
    for (int v = tailStart + tid; v < VOCAB; v += NTHREADS) accum(xrow[v]);
    __syncthreads();
    if (tid == 0) {
      unsigned long long a = acc;
      int b = nb - 1;                                        // descending: big probs first
      for (; b > 0; --b) {
        unsigned long long na = a + bins[b];
        bool cross = strict ? (na > T) : (na >= T);
        if (cross) break;
        a = na;
      }
      *sh_prefix = (prefix << bits) | (unsigned int)b;
      *sh_acc = a;
      *sh_bin = bins[b];
    }
    __syncthreads();
    prefix = *sh_prefix;
    acc = *sh_acc;
    chosen = *sh_bin;
    pb += bits;
    __syncthreads();                                         // before bins re-zeroed
  }
  *outAcc = acc;
  *outBinSum = chosen;
  return prefix;
}

__global__ __launch_bounds__(NTHREADS)
void topp_sample_kernel(const float* __restrict__ logits,
                        const float* __restrict__ uvec,
                        const int* __restrict__ past,
                        float* __restrict__ outTok,
                        float* __restrict__ outProb)
{
  __shared__ unsigned long long ldsPool[POOL_QW];            // counts, later radix bins
  __shared__ float redM[NWAVES];
  __shared__ float redS[NWAVES];
  __shared__ float s_sp[16];
  __shared__ float s_rowM;
  __shared__ float s_Z;
  __shared__ unsigned int sh_prefix;
  __shared__ unsigned long long sh_acc, sh_bin;
  __shared__ int s_tok;

  const int row = blockIdx.x;
  const int tid = threadIdx.x;
  const float* lrow = logits + (size_t)row * VOCAB;
  float* xrow = outProb + (size_t)row * VOCAB;               // staged x; final probs

  // Row stride 50257 = 1 (mod 4): peel `head` scalars so the bulk is 16B-aligned.
  const int head = (4 - (row & 3)) & 3;
  const int nvec = (VOCAB - head) >> 2;
  const int tailStart = head + (nvec << 2);

  // ---- Pass 0: byte-packed occurrence counts of past tokens in LDS ----
  unsigned int* counts = (unsigned int*)ldsPool;
  for (int i = tid; i < CWORDS; i += NTHREADS) counts[i] = 0u;
  __syncthreads();
  if (tid < PAST_LEN) {
    unsigned int t = (unsigned int)past[row * PAST_LEN + tid];
    atomicAdd(&counts[t >> 2], 1u << ((t & 3u) * 8u));       // counts<=64, no carry
  }
  __syncthreads();

  // ---- Pass 1: rep-penalty + temperature, stage x, online softmax ----
  float m = -3.0e38f, s = 0.0f;
  auto stage = [&](float l, int v) {
    unsigned int c = (counts[v >> 2] >> ((v & 3) * 8)) & 0xFFu;
    if (c) {
      float f = exp2f((float)c * 0.26303440583379383f);      // 1.2^c
      l = (l < 0.0f) ? (l * f) : (l / f);
    }
    float x = l * (1.0f / 0.6f);
    float mn = fmaxf(m, x);
    s = s * expf(m - mn) + expf(x - mn);
    m = mn;
    return x;
  };
  for (int v = tid; v < head; v += NTHREADS) xrow[v] = stage(lrow[v], v);
  const v4f* lv = (const v4f*)(lrow + head);
  v4f* xv = (v4f*)(xrow + head);
  for (int i = tid; i < nvec; i += NTHREADS) {
    v4f l4 = __builtin_nontemporal_load(lv + i);             // logits read once: NT
    int v = head + (i << 2);
    v4f x4;
    x4.x = stage(l4.x, v);
    x4.y = stage(l4.y, v + 1);
    x4.z = stage(l4.z, v + 2);
    x4.w = stage(l4.w, v + 3);
    xv[i] = x4;                                              // x re-read 7x: keep RT/L2
  }
  for (int v = tailStart + tid; v < VOCAB; v += NTHREADS) xrow[v] = stage(lrow[v], v);

  for (int off = 16; off > 0; off >>= 1) {                   // wave32 reduction
    float mo = __shfl_xor(m, off, 32);
    float so = __shfl_xor(s, off, 32);
    float mn = fmaxf(m, mo);
    s = s * expf(m - mn) + so * expf(mo - mn);
    m = mn;
  }
  if ((tid & 31) == 0) { redM[tid >> 5] = m; redS[tid >> 5] = s; }
  __syncthreads();
  if (tid == 0) {
    float M = redM[0];
    for (int i = 1; i < NWAVES; ++i) M = fmaxf(M, redM[i]);
    s_rowM = M;
  }
  __syncthreads();
  if (tid < NWAVES) s_sp[tid] = redS[tid] * expf(redM[tid] - s_rowM);
  __syncthreads();

  // ---- WMMA reduction: Z = sum of 16 rescaled per-wave partials ----
  // B = all-ones makes the result invariant to A's lane-layout permutation.
  if (tid < 32) {                                            // wave 0, EXEC all-1s
    const bool hiHalf = tid >= 16;
    v8f acc = {};
    v2f bOnes; bOnes.x = 1.0f; bOnes.y = 1.0f;
    #pragma unroll
    for (int c = 0; c < 4; ++c) {                            // K=4 per op, 4 chained
      const int k0 = 4 * c + (hiHalf ? 2 : 0);
      v2f a; a.x = s_sp[k0]; a.y = s_sp[k0 + 1];
      acc = __builtin_amdgcn_wmma_f32_16x16x4_f32(false, a, false, bOnes,
                                                  (short)0, acc, false, false);
    }
    if (tid == 0) s_Z = acc[0];                              // every D element == Z
  }
  __syncthreads();

  const float rowM = s_rowM;
  unsigned long long* bins = (unsigned long long*)ldsPool;

  // ---- Select 1: top-p crossing key (inclusive cum >= 0.9 * Z) ----
  unsigned long long T0 = (unsigned long long)(0.9 * (double)s_Z * 4294967296.0);
  unsigned long long accK, binK;
  unsigned int kStar = radix_select(xrow, rowM, head, nvec, tailStart, T0, false,
                                    bins, &sh_prefix, &sh_acc, &sh_bin, &accK, &binK);
  // Kept = keys strictly above kStar. If the crossing hit the very top element,
  // force-keep top-1 (reference keep[:,0]=True): include key==kStar itself.
  unsigned long long SkFix;
  if (accK == 0ull) { kStar -= 1u; SkFix = binK; } else { SkFix = accK; }

  // ---- Select 2: inverse-CDF sample point (inclusive cum > u * S_kept) ----
  double Sk = (double)SkFix;
  unsigned long long target = (unsigned long long)((double)uvec[row] * Sk);
  unsigned long long acc2, bin2;
  unsigned int kSamp = radix_select(xrow, rowM, head, nvec, tailStart, target, true,
                                    bins, &sh_prefix, &sh_acc, &sh_bin, &acc2, &bin2);

  // ---- Final pass: renormalized filtered probs + sampled token index ----
  if (tid == 0) s_tok = 0x7FFFFFFF;
  __syncthreads();
  const float SkF = (float)(Sk * (1.0 / 4294967296.0));
  auto finalize = [&](float x, int v) {
    float p = expf(x - rowM);
    unsigned int k = __float_as_uint(p);
    if (k == kSamp) atomicMin(&s_tok, v);                    // first-in-sort tie break
    return (k > kStar) ? (p / SkF) : 0.0f;
  };
  for (int v = tid; v < head; v += NTHREADS) xrow[v] = finalize(xrow[v], v);
  for (int i = tid; i < nvec; i += NTHREADS) {
    v4f x4 = xv[i];
    int v = head + (i << 2);
    v4f o;
    o.x = finalize(x4.x, v);
    o.y = finalize(x4.y, v + 1);
    o.z = finalize(x4.z, v + 2);
    o.w = finalize(x4.w, v + 3);
    __builtin_nontemporal_store(o, xv + i);                  // write-once result: NT
  }
  for (int v = tailStart + tid; v < VOCAB; v += NTHREADS) xrow[v] = finalize(xrow[v], v);
  __syncthreads();
  if (tid == 0) outTok[row] = (float)s_tok;
}

extern "C" void kernel_launch(void* const* d_in, const int* in_sizes, int n_in,
                              void* d_out, int out_size, void* d_ws, size_t ws_size,
                              hipStream_t stream) {
  (void)in_sizes; (void)n_in; (void)out_size; (void)d_ws; (void)ws_size;
  const float* logits = (const float*)d_in[0];
  const float* u      = (const float*)d_in[1];
  const int*   past   = (const int*)d_in[2];
  float* outTok  = (float*)d_out;                            // next_tokens [512] (as float)
  float* outProb = (float*)d_out + BATCH;                    // filt_probs [512, 50257]
  topp_sample_kernel<<<BATCH, NTHREADS, 0, stream>>>(logits, u, past, outTok, outProb);
}